// RecommendationLoss_28295244546057
// MI455X (gfx1250) — compile-verified
//
#include <hip/hip_runtime.h>

typedef float v2f __attribute__((ext_vector_type(2)));
typedef float v8f __attribute__((ext_vector_type(8)));

#define NMAX 8192
#define KPC 16

#define LOG2E 1.4426950408889634f
#define LN2   0.6931471805599453f

// ---------------------------------------------------------------------------
// ws layout (floats): [0]=bpr_loss_sum [1]=pair_count [2]=mse_sum
//                     [3]=valid_count  [4]=entropy_sum
// ---------------------------------------------------------------------------

__global__ void rl_init_kernel(float* ws) {
    if (threadIdx.x == 0) {
        ws[0] = 0.f; ws[1] = 0.f; ws[2] = 0.f; ws[3] = 0.f; ws[4] = 0.f;
    }
}

// Pairwise BPR: one block per 16-row i-tile, 8 waves sweep the j-tiles.
// V_WMMA_F32_16X16X4_F32 produces the difference tiles as rank-2 outer
// products; the pred tile is pre-scaled by -log2e inside the matrix op so
// the element loop is med3 -> v_exp_f32 -> add -> v_log_f32 (softplus).
__global__ void __launch_bounds__(256)
rl_bpr_kernel(const float* __restrict__ pred, const float* __restrict__ targ,
              float* __restrict__ ws, int n) {
    __shared__ float sp[NMAX];
    __shared__ float st[NMAX];
    for (int idx = threadIdx.x; idx < n; idx += 256) {
        sp[idx] = pred[idx];
        st[idx] = targ[idx];
    }
    __syncthreads();

    const int lane = threadIdx.x & 31;
    const int wave = threadIdx.x >> 5;
    const int i0   = blockIdx.x << 4;          // 16-row tile base
    const int ntj  = n >> 4;                   // number of 16-col j tiles

    // A fragments: 16x4 f32. Lanes 0-15: {K=0, K=1} for row M=lane.
    // Lanes 16-31 carry K=2,3 = 0.
    //   pred path: A = [-log2e*p_i | 1],  B = [1 ; +log2e*p_j]
    //     -> D = log2e*(p_j - p_i) = -log2e*diff   (exp2-ready)
    //   targ path: A = [t_i | 1],        B = [1 ; -t_j]
    //     -> D = t_i - t_j                          (mask compare)
    v2f a_p = {0.f, 0.f}, a_t = {0.f, 0.f};
    if (lane < 16) {
        a_p.x = -LOG2E * sp[i0 + lane]; a_p.y = 1.0f;
        a_t.x = st[i0 + lane];          a_t.y = 1.0f;
    }

    float lsum = 0.f;          // sum of masked log2(1 + e^{-diff_clamped})
    float csum = 0.f;          // pair count

    #pragma unroll 2
    for (int jt = wave; jt < ntj; jt += 8) {
        const int j0 = jt << 4;
        v2f b_p = {0.f, 0.f}, b_t = {0.f, 0.f};
        if (lane < 16) {
            b_p.x = 1.0f; b_p.y = LOG2E * sp[j0 + lane];
            b_t.x = 1.0f; b_t.y = -st[j0 + lane];
        }
        v8f z = {};
        v8f d2 = __builtin_amdgcn_wmma_f32_16x16x4_f32(
                     false, a_p, false, b_p, (short)0, z, false, false);
        v8f td = __builtin_amdgcn_wmma_f32_16x16x4_f32(
                     false, a_t, false, b_t, (short)0, z, false, false);

#pragma unroll
        for (int r = 0; r < 8; ++r) {
            float dv = d2[r];                 // -log2e * diff
            float tv = td[r];
            // pair mask: targets strictly ordered + both pred values finite.
            // NaN targets -> tv NaN -> compare false; NaN preds -> dv!=dv.
            bool pm = (tv > 0.0f) && (dv == dv);
            // clamp(diff,-10,10)*(-log2e) == clamp(dv, -10*log2e, 10*log2e)
            float ec = __builtin_amdgcn_fmed3f(dv, -10.0f * LOG2E,
                                                    10.0f * LOG2E);
            float e  = __builtin_amdgcn_exp2f(ec);        // e^{-clamped diff}
            // softplus: -log(sigmoid(diff)+1e-10) ~= log2(1+e) * ln2
            float lg = __builtin_amdgcn_logf(1.0f + e);   // log2(1+e)
            lsum += pm ? lg : 0.0f;
            csum += pm ? 1.0f : 0.0f;
        }
    }

    // wave32 reduction, then one global atomic per wave
#pragma unroll
    for (int off = 16; off > 0; off >>= 1) {
        lsum += __shfl_down(lsum, off, 32);
        csum += __shfl_down(csum, off, 32);
    }
    if (lane == 0) {
        atomicAdd(&ws[0], lsum * LN2);       // -> sum of -ln(sigmoid(diff))
        atomicAdd(&ws[1], csum);
    }
}

__global__ void __launch_bounds__(256)
rl_mse_kernel(const float* __restrict__ pred, const float* __restrict__ targ,
              float* __restrict__ ws, int n) {
    int idx = blockIdx.x * 256 + threadIdx.x;
    float ms = 0.f, vs = 0.f;
    if (idx < n) {
        float p = pred[idx], t = targ[idx];
        bool valid = (p == p) && (t == t);
        if (valid) {
            float df = p - t;
            ms = fminf(df * df, 10.0f);
            vs = 1.0f;
        }
    }
#pragma unroll
    for (int off = 16; off > 0; off >>= 1) {
        ms += __shfl_down(ms, off, 32);
        vs += __shfl_down(vs, off, 32);
    }
    if ((threadIdx.x & 31) == 0) {
        atomicAdd(&ws[2], ms);
        atomicAdd(&ws[3], vs);
    }
}

// One thread per row: softmax over K=16, clamp at 1e-6, entropy.
// exp2/log2 throughout; ln2 base conversion hoisted to the very end.
__global__ void __launch_bounds__(256)
rl_entropy_kernel(const float* __restrict__ pcw, float* __restrict__ ws, int n) {
    int row = blockIdx.x * 256 + threadIdx.x;
    float ent = 0.f;
    if (row < n) {
        const float* x = pcw + (size_t)row * KPC;
        float xv[KPC];
        float m = -3.402823466e38f;
#pragma unroll
        for (int k = 0; k < KPC; ++k) { xv[k] = x[k]; m = fmaxf(m, xv[k]); }
        float sum = 0.f;
#pragma unroll
        for (int k = 0; k < KPC; ++k) {
            xv[k] = __builtin_amdgcn_exp2f((xv[k] - m) * LOG2E);  // e^(x-m)
            sum += xv[k];
        }
        float inv = __builtin_amdgcn_rcpf(sum);
        float e2 = 0.f;
#pragma unroll
        for (int k = 0; k < KPC; ++k) {
            float p = fmaxf(xv[k] * inv, 1e-6f);
            e2 -= p * __builtin_amdgcn_logf(p);    // -p*log2(p)
        }
        ent = e2 * LN2;                             // -> -p*ln(p)
    }
#pragma unroll
    for (int off = 16; off > 0; off >>= 1) ent += __shfl_down(ent, off, 32);
    if ((threadIdx.x & 31) == 0) atomicAdd(&ws[4], ent);
}

__global__ void rl_final_kernel(const float* __restrict__ fa,
                                const float* __restrict__ ws,
                                float* __restrict__ out, int n) {
    if (threadIdx.x != 0 || blockIdx.x != 0) return;
    float mse  = ws[2] / fmaxf(ws[3], 1.0f);
    float bpr  = (ws[1] > 0.f) ? (ws[0] / fmaxf(ws[1], 1.0f)) : 0.f;
    float entm = ws[4] / (float)n;
    float dreg = -0.05f * entm;
    float freg = fmaxf(0.0f, 0.05f - fa[0]);
    out[0] = 1.0f * mse + 0.5f * bpr + 0.1f * dreg + 0.01f * freg;
}

extern "C" void kernel_launch(void* const* d_in, const int* in_sizes, int n_in,
                              void* d_out, int out_size, void* d_ws, size_t ws_size,
                              hipStream_t stream) {
    (void)n_in; (void)out_size; (void)ws_size;
    const float* pred = (const float*)d_in[0];
    const float* targ = (const float*)d_in[1];
    const float* pcw  = (const float*)d_in[2];
    const float* fa   = (const float*)d_in[3];
    float* out = (float*)d_out;
    float* ws  = (float*)d_ws;

    const int n = in_sizes[0];            // 8192
    const int nTiles = n >> 4;            // 512 i-tiles
    const int nBlk1D = (n + 255) / 256;   // 32 blocks for N-sized sweeps

    rl_init_kernel<<<1, 32, 0, stream>>>(ws);
    rl_bpr_kernel<<<nTiles, 256, 0, stream>>>(pred, targ, ws, n);
    rl_mse_kernel<<<nBlk1D, 256, 0, stream>>>(pred, targ, ws, n);
    rl_entropy_kernel<<<nBlk1D, 256, 0, stream>>>(pcw, ws, n);
    rl_final_kernel<<<1, 1, 0, stream>>>(fa, ws, out, n);
}